// HOG_38431367365065
// MI455X (gfx1250) — compile-verified
//
#include <hip/hip_runtime.h>

// HOG on MI455X (gfx1250, wave32).
// Stage 1: per-cell 9-bin histograms via LDS ds_add_f32 atomics (VALU/mem bound part).
// Stage 2: overlapping 2x2 block sum-of-squares as S = P*Csq*P^T using
//          V_WMMA_F32_16X16X4_F32 (exact f32 matrix path).
// Stage 3: normalization out = hist / sqrt(blockSq + eps).

typedef float v2f __attribute__((ext_vector_type(2)));
typedef float v8f __attribute__((ext_vector_type(8)));

#define IMG_H 1024
#define IMG_W 1024
#define CELLS 64   // cells per row/col (1024/16)
#define NB 9
#define PI_F 3.14159265358979323846f

// ---------------------------------------------------------------------------
// Kernel 1: one workgroup per (image, 16-row cell strip).
// 256 threads sweep the 16x1024 strip fully coalesced; LDS atomics build the
// 64 cell histograms; also emit per-cell sum of squared (hist/256) bins.
// ---------------------------------------------------------------------------
__global__ void hog_cell_hist(const float* __restrict__ im,
                              float* __restrict__ hist,   // [img][64][64][9]
                              float* __restrict__ csq) {  // [img][64][64]
  __shared__ float s_hist[CELLS * NB];  // 576 floats
  const int tid = threadIdx.x;
  const int bid = blockIdx.x;
  const int img = bid >> 6;
  const int strip = bid & 63;
  const float GAIN = (float)(9.0 / 3.14159265358979323846);

  for (int i = tid; i < CELLS * NB; i += 256) s_hist[i] = 0.0f;
  __syncthreads();

  const float* base = im + (size_t)img * (IMG_H * IMG_W);
  for (int it = 0; it < 64; ++it) {
    int p = it * 256 + tid;        // 0..16383 within strip
    int r = p >> 10;               // strip-local row 0..15
    int col = p & 1023;
    int gr = strip * 16 + r;       // global row
    const float* rowp = base + (size_t)gr * IMG_W;
    float left  = (col > 0)         ? rowp[col - 1]     : 0.0f;
    float right = (col < IMG_W - 1) ? rowp[col + 1]     : 0.0f;
    float up    = (gr > 0)          ? rowp[col - IMG_W] : 0.0f;
    float down  = (gr < IMG_H - 1)  ? rowp[col + IMG_W] : 0.0f;
    float gx = right - left;
    float gy = down - up;
    float mag = sqrtf(gx * gx + gy * gy);
    float ph = atan2f(gy, gx);
    if (ph < 0.0f) ph += PI_F;          // arctan2 % pi
    if (ph >= PI_F) ph -= PI_F;
    float t = GAIN * ph;
    float fl = floorf(t);
    int li = (int)fl;
    li = li < 0 ? 0 : (li > NB - 1 ? NB - 1 : li);
    float frac = t - fl;                // floor-before-clip, as reference
    int c9 = (col >> 4) * NB;
    atomicAdd(&s_hist[c9 + li], mag * (1.0f - frac));   // ds_add_f32
    int ri = li + 1; if (ri == NB) ri = 0;
    atomicAdd(&s_hist[c9 + ri], mag * frac);
  }
  __syncthreads();

  float* hout = hist + ((size_t)img * CELLS + strip) * (CELLS * NB);
  for (int i = tid; i < CELLS * NB; i += 256)
    hout[i] = s_hist[i] * (1.0f / 256.0f);

  if (tid < CELLS) {
    float s = 0.0f;
    for (int b = 0; b < NB; ++b) {
      float v = s_hist[tid * NB + b] * (1.0f / 256.0f);
      s += v * v;
    }
    csq[((size_t)img * CELLS + strip) * CELLS + tid] = s;
  }
}

// ---------------------------------------------------------------------------
// Kernel 2: block sum-of-squares via WMMA. One workgroup (8 waves) per image.
//   T = Csq * B1   (T[r,j] = Csq[r,j] + Csq[r,j+1])
//   S = P * T      (S[i,j] = T[i,j] + T[i+1,j])
// B1/P are 0/1 band matrices -> exact f32 arithmetic.
// WMMA f32 16x16x4 layouts (wave32):
//   A: M = lane%16, K = vgpr + 2*(lane/16)
//   B: N = lane%16, K = vgpr + 2*(lane/16)   (row-striped, symmetric to A)
//   D: N = lane%16, M = vgpr + 8*(lane/16)
// ---------------------------------------------------------------------------
__global__ void hog_block_pool(const float* __restrict__ csq,
                               float* __restrict__ bsq) {
  __shared__ float s_c[64 * 64];
  __shared__ float s_t[64 * 64];
  const int tid = threadIdx.x;
  const int img = blockIdx.x;
  const int lane = tid & 31;
  const int wave = tid >> 5;
  const int nLo = lane & 15;   // M (for A) or N (for B/D)
  const int hHi = lane >> 4;   // lane-half select

  const float* cin = csq + (size_t)img * 4096;
  for (int i = tid; i < 4096; i += 256) s_c[i] = cin[i];
  __syncthreads();

  // ---- pass 1: T = Csq x B1, B1[k,j] = (k==j) || (k==j+1) ----
  for (int t8 = 0; t8 < 2; ++t8) {
    int tile = wave * 2 + t8;          // 16 tiles cover 64x64
    int r0 = (tile >> 2) << 4;
    int n0 = (tile & 3) << 4;
    v8f acc = {0.f, 0.f, 0.f, 0.f, 0.f, 0.f, 0.f, 0.f};
    int n = n0 + nLo;
    // band: only K in [n0, n0+16] contributes
    for (int k0 = n0; k0 < 64 && k0 <= n0 + 16; k0 += 4) {
      int ka = k0 + 2 * hHi;
      v2f a, b;
      a.x = s_c[(r0 + nLo) * 64 + ka];
      a.y = s_c[(r0 + nLo) * 64 + ka + 1];
      b.x = (ka == n || ka == n + 1) ? 1.0f : 0.0f;
      b.y = ((ka + 1) == n || (ka + 1) == n + 1) ? 1.0f : 0.0f;
      acc = __builtin_amdgcn_wmma_f32_16x16x4_f32(false, a, false, b,
                                                  (short)0, acc, false, false);
    }
    for (int d = 0; d < 8; ++d)
      s_t[(r0 + d + 8 * hHi) * 64 + (n0 + nLo)] = acc[d];
  }
  __syncthreads();

  // ---- pass 2: S = P x T, P[i,k] = (k==i) || (k==i+1) ----
  for (int t8 = 0; t8 < 2; ++t8) {
    int tile = wave * 2 + t8;
    int r0 = (tile >> 2) << 4;
    int n0 = (tile & 3) << 4;
    v8f acc = {0.f, 0.f, 0.f, 0.f, 0.f, 0.f, 0.f, 0.f};
    int m = r0 + nLo;
    // band: only K in [r0, r0+16] contributes
    for (int k0 = r0; k0 < 64 && k0 <= r0 + 16; k0 += 4) {
      int ka = k0 + 2 * hHi;
      v2f a, b;
      a.x = (ka == m || ka == m + 1) ? 1.0f : 0.0f;
      a.y = ((ka + 1) == m || (ka + 1) == m + 1) ? 1.0f : 0.0f;
      b.x = s_t[ka * 64 + n0 + nLo];
      b.y = s_t[(ka + 1) * 64 + n0 + nLo];
      acc = __builtin_amdgcn_wmma_f32_16x16x4_f32(false, a, false, b,
                                                  (short)0, acc, false, false);
    }
    for (int d = 0; d < 8; ++d)
      s_c[(r0 + d + 8 * hHi) * 64 + (n0 + nLo)] = acc[d];  // overwrite with S
  }
  __syncthreads();

  float* bout = bsq + (size_t)img * 4096;
  for (int i = tid; i < 4096; i += 256) bout[i] = s_c[i];
}

// ---------------------------------------------------------------------------
// Kernel 3: out[img,i,j,bi,bj,b] = hist[img,i+bi,j+bj,b] / sqrt(S[i,j]+eps)
// ---------------------------------------------------------------------------
__global__ void hog_normalize(const float* __restrict__ hist,
                              const float* __restrict__ bsq,
                              float* __restrict__ out) {
  const int per = 63 * 63 * 36;  // 142884 outputs per image
  int idx = blockIdx.x * 256 + threadIdx.x;
  if (idx >= per) return;
  int img = blockIdx.y;
  int b = idx % 9;
  int r = idx / 9;
  int bj = r & 1; r >>= 1;
  int bi = r & 1; r >>= 1;
  int j = r % 63;
  int i = r / 63;
  float hval = hist[(size_t)img * (64 * 64 * 9) + (size_t)(i + bi) * (64 * 9)
                    + (j + bj) * 9 + b];
  float s = bsq[(size_t)img * 4096 + i * 64 + j];
  out[(size_t)img * per + idx] = hval / sqrtf(s + 1e-12f);
}

extern "C" void kernel_launch(void* const* d_in, const int* in_sizes, int n_in,
                              void* d_out, int out_size, void* d_ws, size_t ws_size,
                              hipStream_t stream) {
  const float* im = (const float*)d_in[0];
  float* out = (float*)d_out;
  const int nImg = in_sizes[0] / (IMG_H * IMG_W);  // B*C = 24

  // workspace layout: hist | csq | bsq  (~4.3 MB total)
  float* hist = (float*)d_ws;                              // nImg*64*64*9
  float* csq  = hist + (size_t)nImg * 64 * 64 * 9;         // nImg*4096
  float* bsq  = csq  + (size_t)nImg * 4096;                // nImg*4096

  hog_cell_hist<<<dim3(nImg * 64), dim3(256), 0, stream>>>(im, hist, csq);
  hog_block_pool<<<dim3(nImg), dim3(256), 0, stream>>>(csq, bsq);
  dim3 gridN((63 * 63 * 36 + 255) / 256, nImg);
  hog_normalize<<<gridN, dim3(256), 0, stream>>>(hist, bsq, out);
}